// Rips_566935683249
// MI455X (gfx1250) — compile-verified
//
#include <hip/hip_runtime.h>
#include <hip/hip_bf16.h>

#define NPTS 4096
#define DIM  64
#define INF_F __builtin_huge_valf()

typedef __attribute__((ext_vector_type(2))) float v2f;
typedef __attribute__((ext_vector_type(8))) float v8f;

// ---------------------------------------------------------------------------
// Phase 0: squared norms sq[i] = sum_k P[i][k]^2
// ---------------------------------------------------------------------------
__global__ void sqnorm_kernel(const float* __restrict__ P, float* __restrict__ sq) {
    int i = blockIdx.x * blockDim.x + threadIdx.x;
    if (i < NPTS) {
        const float* r = P + (size_t)i * DIM;
        float s = 0.f;
        #pragma unroll
        for (int k = 0; k < DIM; ++k) s += r[k] * r[k];
        sq[i] = s;
    }
}

// ---------------------------------------------------------------------------
// Phase 1: D[i][j] = sqrt(max(sq[i]+sq[j]-2*(P P^T)[i][j], 0)), diag -> 1.0
// One wave32 computes one 16x16 tile via 16 chained v_wmma_f32_16x16x4_f32.
// 4 waves / block, 65536 tiles total -> 16384 blocks of 128 threads.
// ---------------------------------------------------------------------------
__global__ __launch_bounds__(128) void gemm_dist_kernel(
        const float* __restrict__ P, const float* __restrict__ sq,
        float* __restrict__ D) {
    const int lane   = threadIdx.x & 31;
    const int waveId = threadIdx.x >> 5;
    const int tile   = blockIdx.x * 4 + waveId;      // 0 .. 65535
    const int ti     = tile >> 8;                    // row tile (0..255)
    const int tj     = tile & 255;                   // col tile (0..255)
    const int row_base = ti * 16;
    const int col_base = tj * 16;

    // A (16x4 f32) layout: lane%16 = M row; lanes<16 hold K={0,1}, lanes>=16 K={2,3}
    // B (4x16 f32) mirrors with lane%16 = N col. B[k][n] = P[col_base+n][k].
    const int m  = lane & 15;
    const int kh = (lane >> 4) << 1;                 // 0 or 2
    const float* Arow = P + (size_t)(row_base + m) * DIM;
    const float* Brow = P + (size_t)(col_base + m) * DIM;

    v8f c = {};
    #pragma unroll
    for (int kk = 0; kk < DIM; kk += 4) {
        float2 af = *(const float2*)(Arow + kk + kh);
        float2 bf = *(const float2*)(Brow + kk + kh);
        v2f a; a.x = af.x; a.y = af.y;
        v2f b; b.x = bf.x; b.y = bf.y;
        // 8 args: (neg_a, A, neg_b, B, c_mod, C, reuse_a, reuse_b)
        c = __builtin_amdgcn_wmma_f32_16x16x4_f32(
                false, a, false, b, (short)0, c, false, false);
    }

    // C/D layout: VGPR r holds M=r (lanes 0-15) and M=8+r (lanes 16-31); N = lane%16
    const int n     = lane & 15;
    const int mbase = (lane >> 4) * 8;
    const float sqn = sq[col_base + n];
    #pragma unroll
    for (int r = 0; r < 8; ++r) {
        int gi = row_base + mbase + r;
        int gj = col_base + n;
        float d2 = sq[gi] + sqn - 2.0f * c[r];
        d2 = fmaxf(d2, 0.0f);
        if (gi == gj) d2 = 1.0f;                     // safe positive diagonal
        D[(size_t)gi * NPTS + gj] = __builtin_sqrtf(d2);
    }
}

// ---------------------------------------------------------------------------
// Phase 2+3: Prim's MST (single 1024-thread workgroup, dist[] in LDS,
// argmin via packed-u64 wave32 shuffle reduction) then in-LDS bitonic sort.
// Rows of D are L2-resident (64 MB << 192 MB L2).
// ---------------------------------------------------------------------------
__device__ __forceinline__ unsigned long long pack_key(float v, int i) {
    // v >= 0 (distance or +inf) -> float bits are monotone; ties -> lowest index
    return ((unsigned long long)__float_as_uint(v) << 32) | (unsigned)i;
}

__global__ __launch_bounds__(1024) void prim_sort_kernel(
        const float* __restrict__ D, float* __restrict__ out) {
    __shared__ float dist[NPTS];
    __shared__ float sortbuf[NPTS];
    __shared__ unsigned long long wred[32];

    const int t = threadIdx.x;

    // dist = D[0][:], with vertex 0 in tree (dist = inf)
    for (int i = t; i < NPTS; i += 1024) dist[i] = D[i];
    if (t == 0) dist[0] = INF_F;
    __syncthreads();

    for (int it = 0; it < NPTS - 1; ++it) {
        // local min over 4 strided elements
        unsigned long long best = ~0ull;
        #pragma unroll
        for (int s = 0; s < 4; ++s) {
            int i = t + s * 1024;
            unsigned long long key = pack_key(dist[i], i);
            best = key < best ? key : best;
        }
        // wave32 shuffle reduction
        #pragma unroll
        for (int off = 16; off > 0; off >>= 1) {
            unsigned long long o = __shfl_xor(best, off, 32);
            best = o < best ? o : best;
        }
        if ((t & 31) == 0) wred[t >> 5] = best;
        __syncthreads();
        if (t < 32) {                                // exactly wave 0
            unsigned long long b2 = wred[t];
            #pragma unroll
            for (int off = 16; off > 0; off >>= 1) {
                unsigned long long o = __shfl_xor(b2, off, 32);
                b2 = o < b2 ? o : b2;
            }
            if (t == 0) wred[0] = b2;
        }
        __syncthreads();
        best = wred[0];
        const int   j = (int)(best & 0xFFFFFFFFu);
        const float w = __uint_as_float((unsigned)(best >> 32));
        if (t == 0) sortbuf[it] = w;

        // dist = where(in_tree, inf, min(dist, D[j])); in_tree <=> dist==inf
        const float* row = D + (size_t)j * NPTS;
        #pragma unroll
        for (int s = 0; s < 4; ++s) {
            int i = t + s * 1024;
            float dv = dist[i];
            float nv = (dv == INF_F) ? INF_F : fminf(dv, row[i]);
            if (i == j) nv = INF_F;
            dist[i] = nv;
        }
        __syncthreads();
    }

    if (t == 0) sortbuf[NPTS - 1] = INF_F;           // pad to power of two
    __syncthreads();

    // bitonic sort ascending of sortbuf[4096]
    for (int k = 2; k <= NPTS; k <<= 1) {
        for (int jj = k >> 1; jj > 0; jj >>= 1) {
            #pragma unroll
            for (int s = 0; s < 4; ++s) {
                int i = t + s * 1024;
                int p = i ^ jj;
                if (p > i) {
                    float a = sortbuf[i];
                    float b = sortbuf[p];
                    bool up = ((i & k) == 0);
                    if ((a > b) == up) { sortbuf[i] = b; sortbuf[p] = a; }
                }
            }
            __syncthreads();
        }
    }

    for (int i = t; i < NPTS - 1; i += 1024) out[i] = sortbuf[i];
}

// ---------------------------------------------------------------------------
extern "C" void kernel_launch(void* const* d_in, const int* in_sizes, int n_in,
                              void* d_out, int out_size, void* d_ws, size_t ws_size,
                              hipStream_t stream) {
    const float* P   = (const float*)d_in[0];
    float*       out = (float*)d_out;
    float*       D   = (float*)d_ws;                                   // 64 MB
    float*       sq  = (float*)((char*)d_ws + (size_t)NPTS * NPTS * 4); // 16 KB

    sqnorm_kernel<<<NPTS / 256, 256, 0, stream>>>(P, sq);
    gemm_dist_kernel<<<(256 * 256) / 4, 128, 0, stream>>>(P, sq, D);
    prim_sort_kernel<<<1, 1024, 0, stream>>>(D, out);
}